// LocalAttentionLayer_67465346286177
// MI455X (gfx1250) — compile-verified
//
#include <hip/hip_runtime.h>
#include <math.h>
#include <float.h>

typedef __attribute__((ext_vector_type(2))) float v2f;
typedef __attribute__((ext_vector_type(8))) float v8f;

#define B_DIM  2
#define N_SEQ  2048
#define C_DIM  512
#define H_NUM  8
#define D_HEAD 64

// ---------------------------------------------------------------------------
// GEMM: C[m,n] = sum_k A[m,k] * Bt[n,k] (+ bias[n])
// A: [M,K] row-major, Bt: [Nn,K] row-major (computes A @ Bt^T).
// Exact fp32 via V_WMMA_F32_16X16X4_F32, wave32.
//
// Each wave owns a 32x64 output tile: 2 A-frags + 4 B-frags -> 8 wmma per
// k-step (10.7 flop/byte). Fragments are double-buffered in registers so the
// next k-step's 6 global loads are in flight while the current 8 wmma run,
// instead of s_wait_loadcnt 0 before every wmma.
//
// Fragment layouts (CDNA5 ISA 7.12.2, 32-bit operands, wave32):
//   A 16x4 : lane L<16 holds A[m0+L][k..k+1]; lane L>=16 holds A[m0+L-16][k+2..k+3]
//   B 4x16 : lane L<16 holds B[k..k+1][n0+L]; lane L>=16 holds B[k+2..k+3][n0+L-16]
//            with B[k][n] = Bt[n][k]  ->  contiguous float2 from a Bt row.
//   C/D    : VGPR r, lane L -> D[m0 + r + 8*(L/16)][n0 + (L%16)]
// ---------------------------------------------------------------------------
#define WMMA_F32(ACC, Af, Bf) \
    ACC = __builtin_amdgcn_wmma_f32_16x16x4_f32(false, (Af), false, (Bf), (short)0, ACC, false, false)

__global__ __launch_bounds__(256) void gemm_rowrow_wmma_f32(
    const float* __restrict__ A, const float* __restrict__ Bt,
    const float* __restrict__ bias, float* __restrict__ C,
    int M, int Nn, int K)
{
    const int wave = threadIdx.x >> 5;
    const int lane = threadIdx.x & 31;
    const int lm   = lane & 15;
    const int half = lane >> 4;                 // 0 or 1

    const int m0 = (blockIdx.y * 8 + wave) * 32;
    const int n0 = blockIdx.x * 64;
    if (m0 >= M) return;                        // wave-uniform: EXEC stays all-ones

    const float* arow0 = A  + (size_t)(m0 + lm) * K + 2 * half;
    const float* arow1 = arow0 + (size_t)16 * K;
    const float* brow  = Bt + (size_t)(n0 + lm) * K + 2 * half;
    const size_t bs = (size_t)16 * K;

    v8f c00 = {}, c01 = {}, c02 = {}, c03 = {};
    v8f c10 = {}, c11 = {}, c12 = {}, c13 = {};

    // Prologue: fragments for k = 0.
    v2f a0 = *(const v2f*)(arow0);
    v2f a1 = *(const v2f*)(arow1);
    v2f b0 = *(const v2f*)(brow);
    v2f b1 = *(const v2f*)(brow + bs);
    v2f b2 = *(const v2f*)(brow + 2 * bs);
    v2f b3 = *(const v2f*)(brow + 3 * bs);

    #pragma unroll 4
    for (int k = 4; k < K; k += 4) {
        // Issue next k-step's loads first, so they overlap the 8 wmma below.
        v2f na0 = *(const v2f*)(arow0 + k);
        v2f na1 = *(const v2f*)(arow1 + k);
        v2f nb0 = *(const v2f*)(brow + k);
        v2f nb1 = *(const v2f*)(brow + bs + k);
        v2f nb2 = *(const v2f*)(brow + 2 * bs + k);
        v2f nb3 = *(const v2f*)(brow + 3 * bs + k);

        WMMA_F32(c00, a0, b0);  WMMA_F32(c01, a0, b1);
        WMMA_F32(c02, a0, b2);  WMMA_F32(c03, a0, b3);
        WMMA_F32(c10, a1, b0);  WMMA_F32(c11, a1, b1);
        WMMA_F32(c12, a1, b2);  WMMA_F32(c13, a1, b3);

        a0 = na0; a1 = na1;
        b0 = nb0; b1 = nb1; b2 = nb2; b3 = nb3;
    }
    // Epilogue: last k-step.
    WMMA_F32(c00, a0, b0);  WMMA_F32(c01, a0, b1);
    WMMA_F32(c02, a0, b2);  WMMA_F32(c03, a0, b3);
    WMMA_F32(c10, a1, b0);  WMMA_F32(c11, a1, b1);
    WMMA_F32(c12, a1, b2);  WMMA_F32(c13, a1, b3);

    float bs0 = 0.f, bs1 = 0.f, bs2 = 0.f, bs3 = 0.f;
    if (bias) {
        bs0 = bias[n0 + lm];
        bs1 = bias[n0 + 16 + lm];
        bs2 = bias[n0 + 32 + lm];
        bs3 = bias[n0 + 48 + lm];
    }
    #pragma unroll
    for (int r = 0; r < 8; ++r) {
        const int row = m0 + r + 8 * half;
        float* crow = C + (size_t)row * Nn + n0 + lm;
        crow[0]  = c00[r] + bs0;
        crow[16] = c01[r] + bs1;
        crow[32] = c02[r] + bs2;
        crow[48] = c03[r] + bs3;
        float* crow2 = crow + (size_t)16 * Nn;
        crow2[0]  = c10[r] + bs0;
        crow2[16] = c11[r] + bs1;
        crow2[32] = c12[r] + bs2;
        crow2[48] = c13[r] + bs3;
    }
}

// ---------------------------------------------------------------------------
// Banded dual-window attention.
// Masks are dist<=4 (ws=9) and dist<=8 (ws=17): only a 17-wide diagonal band
// of the score matrix survives the softmax, so compute just the band.
// One wave32 per (b, h, n) query. Lane L owns head dims L and L+32.
// qkv layout: [B, N, 3, H, D] row-major (exactly the reference reshape).
// attn_out layout: [B, N, H*D] = the reference's transpose(0,2,1,3).reshape.
// ---------------------------------------------------------------------------
__global__ __launch_bounds__(256) void banded_attn_kernel(
    const float* __restrict__ qkv, float* __restrict__ attn_out)
{
    const int lane = threadIdx.x & 31;
    const int wid  = blockIdx.x * 8 + (threadIdx.x >> 5); // 0 .. B*H*N-1
    const int n = wid & (N_SEQ - 1);
    const int h = (wid >> 11) & (H_NUM - 1);              // N_SEQ = 2^11
    const int b = wid >> 14;                              // H*N  = 2^14

    const size_t rowstride = (size_t)3 * C_DIM;           // floats per (b, n)
    const float* qp = qkv + (size_t)(b * N_SEQ + n) * rowstride + h * D_HEAD;
    const float qx = qp[lane];
    const float qy = qp[lane + 32];

    const int j0 = n - 8;
    float s[17];
    #pragma unroll
    for (int r = 0; r < 17; ++r) {
        const int j = j0 + r;
        float sc = -FLT_MAX;
        if (j >= 0 && j < N_SEQ) {                        // wave-uniform branch
            const float* kp = qkv + (size_t)(b * N_SEQ + j) * rowstride
                              + C_DIM + h * D_HEAD;
            float p = qx * kp[lane] + qy * kp[lane + 32];
            #pragma unroll
            for (int off = 16; off > 0; off >>= 1)
                p += __shfl_xor(p, off, 32);              // wave32 reduction
            sc = p * 0.125f;                              // scale = D^-0.5 = 1/8
        }
        s[r] = sc;
    }

    // Dual softmax: window 17 uses all valid r; window 9 uses r in [4,12].
    float m8 = -FLT_MAX, m4 = -FLT_MAX;
    #pragma unroll
    for (int r = 0; r < 17; ++r) {
        m8 = fmaxf(m8, s[r]);
        if (r >= 4 && r <= 12) m4 = fmaxf(m4, s[r]);
    }
    float z8 = 0.f, z4 = 0.f;
    #pragma unroll
    for (int r = 0; r < 17; ++r) {
        z8 += expf(s[r] - m8);                            // invalid -> exp(-inf)=0
        if (r >= 4 && r <= 12) z4 += expf(s[r] - m4);
    }
    const float inv8 = 1.f / z8;
    const float inv4 = 1.f / z4;

    float ax = 0.f, ay = 0.f;
    #pragma unroll
    for (int r = 0; r < 17; ++r) {
        const int j = j0 + r;
        if (j >= 0 && j < N_SEQ) {
            float w = expf(s[r] - m8) * inv8;
            if (r >= 4 && r <= 12) w += expf(s[r] - m4) * inv4;
            w *= 0.5f;                                    // average of 2 windows
            const float* vp = qkv + (size_t)(b * N_SEQ + j) * rowstride
                              + 2 * C_DIM + h * D_HEAD;
            ax += w * vp[lane];
            ay += w * vp[lane + 32];
        }
    }

    float* op = attn_out + (size_t)(b * N_SEQ + n) * C_DIM + h * D_HEAD;
    op[lane]      = ax;
    op[lane + 32] = ay;
}

// ---------------------------------------------------------------------------
extern "C" void kernel_launch(void* const* d_in, const int* in_sizes, int n_in,
                              void* d_out, int out_size, void* d_ws, size_t ws_size,
                              hipStream_t stream)
{
    (void)in_sizes; (void)n_in; (void)out_size; (void)ws_size;
    const float* x      = (const float*)d_in[0]; // [B, N, C]
    const float* qkv_w  = (const float*)d_in[1]; // [3C, C]
    const float* proj_w = (const float*)d_in[2]; // [C, C]
    const float* proj_b = (const float*)d_in[3]; // [C]
    float* out = (float*)d_out;                  // [B, N, C]

    const int M  = B_DIM * N_SEQ;                // 4096
    const int K  = C_DIM;                        // 512
    const int N3 = 3 * C_DIM;                    // 1536

    float* qkv  = (float*)d_ws;                  // [M, 3C]  = 25.2 MB
    float* attn = qkv + (size_t)M * N3;          // [M, C]   =  8.4 MB

    dim3 blk(256);

    // 1) QKV projection: qkv = x @ qkv_w^T        (M=4096, Nn=1536, K=512)
    //    block = 8 waves x (32x64) = 256 rows x 64 cols
    gemm_rowrow_wmma_f32<<<dim3(N3 / 64, M / 256), blk, 0, stream>>>(
        x, qkv_w, nullptr, qkv, M, N3, K);

    // 2) Banded dual-window attention -> attn [M, C]
    const int n_waves = B_DIM * H_NUM * N_SEQ;   // 32768 queries
    banded_attn_kernel<<<dim3(n_waves / 8), blk, 0, stream>>>(qkv, attn);

    // 3) Output projection: out = attn @ proj_w^T + proj_b  (Nn=512, K=512)
    gemm_rowrow_wmma_f32<<<dim3(C_DIM / 64, M / 256), blk, 0, stream>>>(
        attn, proj_w, proj_b, out, M, C_DIM, K);
}